// E8Simplifier_54812372632388
// MI455X (gfx1250) — compile-verified
//
#include <hip/hip_runtime.h>
#include <hip/hip_bf16.h>
#include <math.h>

// ---------------------------------------------------------------------------
// 2-layer GAT (PyG GATConv semantics) for MI455X / gfx1250.
//  L1: x[50000,8] @ W1[8,128] via V_WMMA_F32_16X16X4_F32, per-head attention,
//      segment softmax over dst (3 edge passes), ELU.
//  L2: matvec to scalar, scalar attention softmax, sigmoid.
// ---------------------------------------------------------------------------

typedef float v2f __attribute__((ext_vector_type(2)));
typedef float v8f __attribute__((ext_vector_type(8)));

#define NNODES 50000
#define NEG_SLOPE 0.2f
#define SEPS 1e-16f

__device__ __forceinline__ unsigned fmap(float f) {
  // order-preserving float -> uint map for atomicMax
  unsigned u = __float_as_uint(f);
  return (u & 0x80000000u) ? ~u : (u | 0x80000000u);
}
__device__ __forceinline__ float funmap(unsigned u) {
  return (u & 0x80000000u) ? __uint_as_float(u & 0x7FFFFFFFu)
                           : __uint_as_float(~u);
}
__device__ __forceinline__ float lrelu(float a) {
  return a > 0.0f ? a : a * NEG_SLOPE;
}

// ---------------------------------------------------------------------------
// workspace zeroing (grid-stride)
// ---------------------------------------------------------------------------
__global__ void k_zero(float* __restrict__ p, long n) {
  long i = (long)blockIdx.x * blockDim.x + threadIdx.x;
  long stride = (long)gridDim.x * blockDim.x;
  for (; i < n; i += stride) p[i] = 0.0f;
}

// ---------------------------------------------------------------------------
// Layer-1 GEMM: xl = x @ W1   (x:[N,8], W1:[8,128], xl:[N,128])
// One wave (32 lanes) per 16-row tile; loop over 8 col tiles of 16.
// K=8 handled as two chained V_WMMA_F32_16X16X4_F32 (K=4 each).
// A 16x4 f32 layout: lanes 0-15 -> M=lane, K={0,1}; lanes 16-31 -> K={2,3}.
// B 4x16 f32 layout: lanes 0-15 -> N=lane, K rows {0,1}; lanes 16-31 -> {2,3}.
// D 16x16 f32: vgpr i: lanes 0-15 -> (M=i, N=lane); lanes 16-31 -> (M=i+8).
// ---------------------------------------------------------------------------
__global__ __launch_bounds__(32) void k_l1_gemm(const float* __restrict__ x,
                                                const float* __restrict__ W1,
                                                float* __restrict__ xl) {
  const int tile = blockIdx.x;          // 0..3124 (50000/16 tiles exactly)
  const int lane = threadIdx.x;         // 0..31
  const int m = lane & 15;
  const int hi = lane >> 4;             // 0 -> K 0/1 (or 4/5), 1 -> K 2/3 (6/7)
  const int row0 = tile * 16;

  const float* xr = x + (long)(row0 + m) * 8;
  v2f a0, a1;
  a0.x = xr[hi * 2 + 0];
  a0.y = xr[hi * 2 + 1];
  a1.x = xr[4 + hi * 2 + 0];
  a1.y = xr[4 + hi * 2 + 1];

  for (int t = 0; t < 8; ++t) {
    const int col = t * 16 + m;
    v2f b0, b1;
    b0.x = W1[(hi * 2 + 0) * 128 + col];
    b0.y = W1[(hi * 2 + 1) * 128 + col];
    b1.x = W1[(4 + hi * 2 + 0) * 128 + col];
    b1.y = W1[(4 + hi * 2 + 1) * 128 + col];

    v8f c = {};
    c = __builtin_amdgcn_wmma_f32_16x16x4_f32(false, a0, false, b0,
                                              (short)0, c, false, false);
    c = __builtin_amdgcn_wmma_f32_16x16x4_f32(false, a1, false, b1,
                                              (short)0, c, false, false);
#pragma unroll
    for (int i = 0; i < 8; ++i) {
      xl[(long)(row0 + hi * 8 + i) * 128 + col] = c[i];
    }
  }
}

// ---------------------------------------------------------------------------
// Per-node attention logits: a_src[n,h] = <xl[n,h,:], att_src[h,:]>, same dst.
// One thread per (n,h).
// ---------------------------------------------------------------------------
__global__ void k_l1_att(const float* __restrict__ xl,
                         const float* __restrict__ att_src,
                         const float* __restrict__ att_dst,
                         float* __restrict__ a_src, float* __restrict__ a_dst,
                         int total) {
  int idx = blockIdx.x * blockDim.x + threadIdx.x;
  if (idx >= total) return;
  const int h = idx & 7;
  const float* v = xl + (long)(idx >> 3) * 128 + h * 16;
  const float* as = att_src + h * 16;
  const float* ad = att_dst + h * 16;
  float s = 0.0f, d = 0.0f;
#pragma unroll
  for (int c = 0; c < 16; ++c) {
    s = fmaf(v[c], as[c], s);
    d = fmaf(v[c], ad[c], d);
  }
  a_src[idx] = s;
  a_dst[idx] = d;
}

// ---------------------------------------------------------------------------
// Edge pass 1 (L1): segment max of leaky-relu(alpha) over dst. Thread/edge.
// ---------------------------------------------------------------------------
__global__ void k_edge1_max(const int* __restrict__ ei, int E, int N,
                            const float* __restrict__ a_src,
                            const float* __restrict__ a_dst,
                            unsigned* __restrict__ mmax) {
  int e = blockIdx.x * blockDim.x + threadIdx.x;
  if (e >= E + N) return;
  int s, d;
  if (e < E) { s = ei[e]; d = ei[E + e]; } else { s = d = e - E; }
#pragma unroll
  for (int h = 0; h < 8; ++h) {
    float al = lrelu(a_src[(long)s * 8 + h] + a_dst[(long)d * 8 + h]);
    atomicMax(&mmax[(long)d * 8 + h], fmap(al));
  }
}

// ---------------------------------------------------------------------------
// Edge pass 2 (L1): segment sum of exp(alpha - m[dst]). Thread/edge.
// ---------------------------------------------------------------------------
__global__ void k_edge1_sum(const int* __restrict__ ei, int E, int N,
                            const float* __restrict__ a_src,
                            const float* __restrict__ a_dst,
                            const unsigned* __restrict__ mmax,
                            float* __restrict__ ssum) {
  int e = blockIdx.x * blockDim.x + threadIdx.x;
  if (e >= E + N) return;
  int s, d;
  if (e < E) { s = ei[e]; d = ei[E + e]; } else { s = d = e - E; }
#pragma unroll
  for (int h = 0; h < 8; ++h) {
    float al = lrelu(a_src[(long)s * 8 + h] + a_dst[(long)d * 8 + h]);
    float ex = __expf(al - funmap(mmax[(long)d * 8 + h]));
    atomicAdd(&ssum[(long)d * 8 + h], ex);
  }
}

// ---------------------------------------------------------------------------
// Edge pass 3 (L1): out[dst,h,:] += xl[src,h,:] * softmax-coef. Thread/(e,h).
// ---------------------------------------------------------------------------
__global__ void k_edge1_agg(const int* __restrict__ ei, int E, int N,
                            const float* __restrict__ a_src,
                            const float* __restrict__ a_dst,
                            const unsigned* __restrict__ mmax,
                            const float* __restrict__ ssum,
                            const float* __restrict__ xl,
                            float* __restrict__ acc1) {
  int idx = blockIdx.x * blockDim.x + threadIdx.x;
  int total = (E + N) * 8;
  if (idx >= total) return;
  const int e = idx >> 3;
  const int h = idx & 7;
  int s, d;
  if (e < E) { s = ei[e]; d = ei[E + e]; } else { s = d = e - E; }
  float al = lrelu(a_src[(long)s * 8 + h] + a_dst[(long)d * 8 + h]);
  float ex = __expf(al - funmap(mmax[(long)d * 8 + h]));
  float coef = ex / (ssum[(long)d * 8 + h] + SEPS);
  const float* xv = xl + (long)s * 128 + h * 16;
  __builtin_prefetch(xv, 0, 0);
  float* av = acc1 + (long)d * 128 + h * 16;
#pragma unroll
  for (int c = 0; c < 16; ++c) atomicAdd(&av[c], xv[c] * coef);
}

// ---------------------------------------------------------------------------
// L1 finalize: h1 = ELU(acc1 + b1)   (h1 overwrites xl buffer)
// ---------------------------------------------------------------------------
__global__ void k_l1_final(const float* __restrict__ acc1,
                           const float* __restrict__ b1,
                           float* __restrict__ h1, long total) {
  long i = (long)blockIdx.x * blockDim.x + threadIdx.x;
  if (i >= total) return;
  float v = acc1[i] + b1[i & 127];
  h1[i] = v > 0.0f ? v : (__expf(v) - 1.0f);
}

// ---------------------------------------------------------------------------
// L2 linear: xl2[n] = <h1[n,:], W2[:,0]>
// ---------------------------------------------------------------------------
__global__ void k_l2_lin(const float* __restrict__ h1,
                         const float* __restrict__ W2,
                         float* __restrict__ xl2, int N) {
  int n = blockIdx.x * blockDim.x + threadIdx.x;
  if (n >= N) return;
  const float* hv = h1 + (long)n * 128;
  float s = 0.0f;
#pragma unroll 8
  for (int c = 0; c < 128; ++c) s = fmaf(hv[c], W2[c], s);
  xl2[n] = s;
}

// ---------------------------------------------------------------------------
// L2 edge passes (heads=1, out_ch=1; attention scalars from device memory)
// ---------------------------------------------------------------------------
__global__ void k_edge2_max(const int* __restrict__ ei, int E, int N,
                            const float* __restrict__ xl2,
                            const float* __restrict__ pas,
                            const float* __restrict__ pad,
                            unsigned* __restrict__ m2) {
  int e = blockIdx.x * blockDim.x + threadIdx.x;
  if (e >= E + N) return;
  int s, d;
  if (e < E) { s = ei[e]; d = ei[E + e]; } else { s = d = e - E; }
  float al = lrelu(xl2[s] * pas[0] + xl2[d] * pad[0]);
  atomicMax(&m2[d], fmap(al));
}

__global__ void k_edge2_sum(const int* __restrict__ ei, int E, int N,
                            const float* __restrict__ xl2,
                            const float* __restrict__ pas,
                            const float* __restrict__ pad,
                            const unsigned* __restrict__ m2,
                            float* __restrict__ s2) {
  int e = blockIdx.x * blockDim.x + threadIdx.x;
  if (e >= E + N) return;
  int s, d;
  if (e < E) { s = ei[e]; d = ei[E + e]; } else { s = d = e - E; }
  float al = lrelu(xl2[s] * pas[0] + xl2[d] * pad[0]);
  atomicAdd(&s2[d], __expf(al - funmap(m2[d])));
}

__global__ void k_edge2_agg(const int* __restrict__ ei, int E, int N,
                            const float* __restrict__ xl2,
                            const float* __restrict__ pas,
                            const float* __restrict__ pad,
                            const unsigned* __restrict__ m2,
                            const float* __restrict__ s2,
                            float* __restrict__ acc2) {
  int e = blockIdx.x * blockDim.x + threadIdx.x;
  if (e >= E + N) return;
  int s, d;
  if (e < E) { s = ei[e]; d = ei[E + e]; } else { s = d = e - E; }
  float al = lrelu(xl2[s] * pas[0] + xl2[d] * pad[0]);
  float ex = __expf(al - funmap(m2[d]));
  float coef = ex / (s2[d] + SEPS);
  atomicAdd(&acc2[d], xl2[s] * coef);
}

__global__ void k_final(const float* __restrict__ acc2,
                        const float* __restrict__ b2,
                        float* __restrict__ out, int N) {
  int n = blockIdx.x * blockDim.x + threadIdx.x;
  if (n >= N) return;
  float v = acc2[n] + b2[0];           // heads=1, mean == identity
  out[n] = 1.0f / (1.0f + __expf(-v)); // sigmoid
}

// ---------------------------------------------------------------------------
extern "C" void kernel_launch(void* const* d_in, const int* in_sizes, int n_in,
                              void* d_out, int out_size, void* d_ws,
                              size_t ws_size, hipStream_t stream) {
  const float* x   = (const float*)d_in[0];
  const int*   ei  = (const int*)d_in[1];   // [2,E] flat: src then dst
  const float* W1  = (const float*)d_in[2];
  const float* as1 = (const float*)d_in[3];
  const float* ad1 = (const float*)d_in[4];
  const float* b1  = (const float*)d_in[5];
  const float* W2  = (const float*)d_in[6];
  const float* as2 = (const float*)d_in[7];
  const float* ad2 = (const float*)d_in[8];
  const float* b2  = (const float*)d_in[9];
  float* out = (float*)d_out;

  const int N = NNODES;
  const int E = in_sizes[1] / 2;
  const int ET = E + N;  // edges + self-loops

  // workspace layout (float units)
  float* ws = (float*)d_ws;
  float*    xl    = ws;                      // [N*128]  (reused as h1)
  float*    a_src = ws + 6400000;            // [N*8]
  float*    a_dst = ws + 6800000;            // [N*8]
  unsigned* mmax  = (unsigned*)(ws + 7200000); // [N*8]
  float*    ssum  = ws + 7600000;            // [N*8]
  float*    acc1  = ws + 8000000;            // [N*128]
  float*    xl2   = ws + 14400000;           // [N]
  unsigned* m2    = (unsigned*)(ws + 14450000); // [N]
  float*    s2    = ws + 14500000;           // [N]
  float*    acc2  = ws + 14550000;           // [N]

  const int B = 256;

  // zero accumulators/max/sum tables: [7.2M, 14.6M)
  k_zero<<<2048, B, 0, stream>>>(ws + 7200000, 7400000L);

  // Layer 1
  k_l1_gemm<<<N / 16, 32, 0, stream>>>(x, W1, xl);
  k_l1_att<<<(N * 8 + B - 1) / B, B, 0, stream>>>(xl, as1, ad1, a_src, a_dst,
                                                  N * 8);
  k_edge1_max<<<(ET + B - 1) / B, B, 0, stream>>>(ei, E, N, a_src, a_dst, mmax);
  k_edge1_sum<<<(ET + B - 1) / B, B, 0, stream>>>(ei, E, N, a_src, a_dst, mmax,
                                                  ssum);
  k_edge1_agg<<<(ET * 8 + B - 1) / B, B, 0, stream>>>(ei, E, N, a_src, a_dst,
                                                      mmax, ssum, xl, acc1);
  k_l1_final<<<(int)(((long)N * 128 + B - 1) / B), B, 0, stream>>>(
      acc1, b1, xl, (long)N * 128);  // h1 overwrites xl

  // Layer 2
  k_l2_lin<<<(N + B - 1) / B, B, 0, stream>>>(xl, W2, xl2, N);
  k_edge2_max<<<(ET + B - 1) / B, B, 0, stream>>>(ei, E, N, xl2, as2, ad2, m2);
  k_edge2_sum<<<(ET + B - 1) / B, B, 0, stream>>>(ei, E, N, xl2, as2, ad2, m2,
                                                  s2);
  k_edge2_agg<<<(ET + B - 1) / B, B, 0, stream>>>(ei, E, N, xl2, as2, ad2, m2,
                                                  s2, acc2);
  k_final<<<(N + B - 1) / B, B, 0, stream>>>(acc2, b2, out, N);
}